// SRGNN_30485677867451
// MI455X (gfx1250) — compile-verified
//
#include <hip/hip_runtime.h>
#include <hip/hip_bf16.h>

typedef __attribute__((ext_vector_type(2))) float v2f;
typedef __attribute__((ext_vector_type(8))) float v8f;

#define EMBED 64

// --- degree: init to 1.0 (self loop) ---
__global__ void k_deg_init(float* __restrict__ deg, int n) {
  int i = blockIdx.x * blockDim.x + threadIdx.x;
  if (i < n) deg[i] = 1.0f;
}

// --- degree: accumulate in-degree over dst ---
__global__ void k_deg_accum(const int* __restrict__ dst, float* __restrict__ deg, int E) {
  int e = blockIdx.x * blockDim.x + threadIdx.x;
  if (e < E) atomicAdd(&deg[dst[e]], 1.0f);
}

// --- dinv = rsqrt(deg), in place ---
__global__ void k_rsqrt(float* __restrict__ deg, int n) {
  int i = blockIdx.x * blockDim.x + threadIdx.x;
  if (i < n) deg[i] = rsqrtf(deg[i]);
}

// --- h2 = emb[x] @ W^T via fp32 WMMA (16x16 tile per wave, K=64 in steps of 4) ---
__global__ void k_gemm_wmma(const int* __restrict__ x, const float* __restrict__ emb,
                            const float* __restrict__ W, float* __restrict__ h2, int n) {
  const int wave = (blockIdx.x * blockDim.x + threadIdx.x) >> 5;
  const int lane = threadIdx.x & 31;
  const int tile_m = wave >> 2;       // 4 column-tiles per row block (64/16)
  const int tile_n = wave & 3;
  const int i0 = tile_m * 16;
  if (i0 >= n) return;                // wave-uniform: EXEC stays all-1s for WMMA
  const int j0 = tile_n * 16;
  const int half = lane >> 4;         // 0: K={0,1}; 1: K={2,3} (per 32-bit A layout)
  const int l = lane & 15;

  int ridx = i0 + l; if (ridx >= n) ridx = n - 1;   // defensive clamp
  const int row = x[ridx];                          // embedding gather (identity here)
  const float* __restrict__ arow = emb + (size_t)row * EMBED + half * 2;
  // B[k][nn] = W[j0+nn][k] (since h @ W^T); same lane striping as A with M<->N
  const float* __restrict__ brow = W + (size_t)(j0 + l) * EMBED + half * 2;

  v8f acc = {};
#pragma unroll
  for (int k = 0; k < EMBED; k += 4) {
    v2f a;  a.x  = arow[k]; a.y  = arow[k + 1];
    v2f bm; bm.x = brow[k]; bm.y = brow[k + 1];
    // 8 args: (neg_a, A, neg_b, B, c_mod, C, reuse_a, reuse_b)
    acc = __builtin_amdgcn_wmma_f32_16x16x4_f32(false, a, false, bm, (short)0, acc,
                                                false, false);
  }

  // C/D layout: lanes 0-15 VGPR r -> M=r, N=lane; lanes 16-31 VGPR r -> M=8+r, N=lane-16
#pragma unroll
  for (int r = 0; r < 8; ++r) {
    int m = i0 + half * 8 + r;
    if (m < n) h2[(size_t)m * EMBED + (j0 + l)] = acc[r];
  }
}

// --- out = h2 * dinv^2 + b  (self-loop term + bias; also initializes poisoned d_out) ---
__global__ void k_out_init(const float* __restrict__ h2, const float* __restrict__ dinv,
                           const float* __restrict__ b, float* __restrict__ out, int n) {
  int t = blockIdx.x * blockDim.x + threadIdx.x;
  if (t >= n * EMBED) return;
  int i = t >> 6, c = t & 63;
  float di = dinv[i];
  out[t] = h2[t] * di * di + b[c];
}

// --- edge scatter: out[dst] += h2[src] * dinv[src]*dinv[dst] ---
__global__ void k_scatter(const float* __restrict__ h2, const int* __restrict__ ei,
                          const float* __restrict__ dinv, float* __restrict__ out, int E) {
  int t = blockIdx.x * blockDim.x + threadIdx.x;
  if (t >= E * EMBED) return;         // 1.2M*64 = 76.8M < 2^31
  int e = t >> 6, c = t & 63;
  int s = ei[e];
  int d = ei[E + e];
  float norm = dinv[s] * dinv[d];
  atomicAdd(&out[(size_t)d * EMBED + c], h2[(size_t)s * EMBED + c] * norm);
}

extern "C" void kernel_launch(void* const* d_in, const int* in_sizes, int n_in,
                              void* d_out, int out_size, void* d_ws, size_t ws_size,
                              hipStream_t stream) {
  const int*   x   = (const int*)d_in[0];
  const int*   ei  = (const int*)d_in[1];   // [2, E] row-major: src = ei[0..E), dst = ei[E..2E)
  const float* emb = (const float*)d_in[2];
  const float* W   = (const float*)d_in[3];
  const float* b   = (const float*)d_in[4];
  float* out = (float*)d_out;

  const int N = in_sizes[0];
  const int E = in_sizes[1] / 2;

  // workspace: dinv (N floats) | h2 (N*64 floats)
  float* dinv = (float*)d_ws;
  size_t off = (((size_t)N * sizeof(float) + 255) / 256) * 256;
  float* h2 = (float*)((char*)d_ws + off);

  k_deg_init <<<(N + 255) / 256, 256, 0, stream>>>(dinv, N);
  k_deg_accum<<<(E + 255) / 256, 256, 0, stream>>>(ei + E, dinv, E);
  k_rsqrt    <<<(N + 255) / 256, 256, 0, stream>>>(dinv, N);

  const int waves  = ((N + 15) / 16) * 4;   // 16x16 tile per wave, 4 col-tiles
  const int blocks = (waves + 7) / 8;       // 8 waves (256 threads) per block
  k_gemm_wmma<<<blocks, 256, 0, stream>>>(x, emb, W, h2, N);

  const long long out_elems = (long long)N * EMBED;
  k_out_init<<<(unsigned)((out_elems + 255) / 256), 256, 0, stream>>>(h2, dinv, b, out, N);

  const long long edge_elems = (long long)E * EMBED;
  k_scatter<<<(unsigned)((edge_elems + 255) / 256), 256, 0, stream>>>(h2, ei, dinv, out, E);
}